// TimestepNorm_69492570849449
// MI455X (gfx1250) — compile-verified
//
#include <hip/hip_runtime.h>
#include <stdint.h>

// Problem constants (match reference).
#define T_DIM 8192
#define B_DIM 8
#define D_DIM 512
#define NCHUNK 32
#define CHUNK (T_DIM / NCHUNK)   // 256 timesteps per chunk
#define PF 8                     // async prefetch depth
#define RING 16                  // 2*PF slots -> no LDS WAR hazard with async writes
#define TPB 128                  // each thread owns 4 consecutive features
#define PRIOR_COUNT_F 2.0f
#define EPS_F 1e-5f

// ---- CDNA5 async-DMA-to-LDS helpers (gfx1250) ------------------------------
__device__ __forceinline__ void async_load_b128(uint32_t lds_off, const float* gptr) {
  // VDST = per-lane LDS byte address, VADDR = 64-bit global address (GV mode).
  asm volatile("global_load_async_to_lds_b128 %0, %1, off"
               :: "v"(lds_off), "v"(gptr)
               : "memory");
}
#define S_WAIT_ASYNC_LE7() asm volatile("s_wait_asynccnt 7" ::: "memory")
#define S_WAIT_ASYNC_0()   asm volatile("s_wait_asynccnt 0" ::: "memory")

__device__ __forceinline__ uint32_t lds_addr_of(const void* p) {
  // Low 32 bits of the generic pointer to a __shared__ object are the LDS offset.
  return (uint32_t)(uintptr_t)p;
}

// ---- Phase 1: chunk-local Welford stats ------------------------------------
__global__ __launch_bounds__(TPB) void tsnorm_ph1(
    const float* __restrict__ x, const unsigned char* __restrict__ mask,
    float* __restrict__ lm, float* __restrict__ lM2, float* __restrict__ lcnt) {
  __shared__ __align__(16) float ring[RING][D_DIM];
  const int b = blockIdx.x / NCHUNK;
  const int c = blockIdx.x % NCHUNK;
  const int tid = threadIdx.x;
  const int d4 = tid * 4;
  const int t0 = c * CHUNK;
  const size_t trow = (size_t)B_DIM * D_DIM;
  const float* xbase = x + (size_t)t0 * trow + (size_t)b * D_DIM + d4;
  const unsigned char* mbase = mask + (size_t)b * T_DIM + t0;

  // Prologue: prime PF async loads.
  for (int i = 0; i < PF; ++i)
    async_load_b128(lds_addr_of(&ring[i][d4]), xbase + (size_t)i * trow);

  float m0 = 0.f, m1 = 0.f, m2 = 0.f, m3 = 0.f;
  float s0 = 0.f, s1 = 0.f, s2 = 0.f, s3 = 0.f;
  float n = 0.f;

  auto body = [&](int i) {
    const float4 v = *reinterpret_cast<const float4*>(&ring[i % RING][d4]);
    const float valid = mbase[i] ? 0.f : 1.f;
    const float n_new = n + valid;
    const float coef = (valid > 0.f) ? (valid / n_new) : 0.f;
    float d;
    d = v.x - m0; m0 += coef * d; s0 += valid * d * (v.x - m0);
    d = v.y - m1; m1 += coef * d; s1 += valid * d * (v.y - m1);
    d = v.z - m2; m2 += coef * d; s2 += valid * d * (v.z - m2);
    d = v.w - m3; m3 += coef * d; s3 += valid * d * (v.w - m3);
    n = n_new;
  };

  for (int i = 0; i < CHUNK - PF; ++i) {
    S_WAIT_ASYNC_LE7();          // oldest of the PF in-flight loads is done
    body(i);
    const int j = i + PF;
    async_load_b128(lds_addr_of(&ring[j % RING][d4]), xbase + (size_t)j * trow);
  }
  S_WAIT_ASYNC_0();
  for (int i = CHUNK - PF; i < CHUNK; ++i) body(i);

  const size_t o = ((size_t)b * NCHUNK + c) * D_DIM + d4;
  lm[o + 0] = m0; lm[o + 1] = m1; lm[o + 2] = m2; lm[o + 3] = m3;
  lM2[o + 0] = s0; lM2[o + 1] = s1; lM2[o + 2] = s2; lM2[o + 3] = s3;
  if (tid == 0) lcnt[b * NCHUNK + c] = n;
}

// ---- Phase 2: exclusive combine-scan over chunks (Chan merge) --------------
__global__ __launch_bounds__(D_DIM) void tsnorm_ph2(
    const float* __restrict__ prior_mean, const float* __restrict__ prior_logv,
    const float* __restrict__ lm, const float* __restrict__ lM2,
    const float* __restrict__ lcnt,
    float* __restrict__ cm, float* __restrict__ cM2, float* __restrict__ ccnt) {
  const int b = blockIdx.x;
  const int d = threadIdx.x;
  float n  = PRIOR_COUNT_F;
  float m  = prior_mean[d];
  float M2 = PRIOR_COUNT_F * __expf(prior_logv[d]);
  for (int c = 0; c < NCHUNK; ++c) {
    const size_t o = ((size_t)b * NCHUNK + c) * D_DIM + d;
    cm[o] = m; cM2[o] = M2;
    if (d == 0) ccnt[b * NCHUNK + c] = n;
    const float n2  = lcnt[b * NCHUNK + c];
    const float mu2 = lm[o];
    const float s2  = lM2[o];
    if (n2 > 0.f) {
      const float nn  = n + n2;
      const float dlt = mu2 - m;
      m  += dlt * (n2 / nn);
      M2 += s2 + dlt * dlt * (n * n2 / nn);
      n = nn;
    }
  }
}

// ---- Phase 3: replay chunk from carried-in stats, emit normalized y --------
__global__ __launch_bounds__(TPB) void tsnorm_ph3(
    const float* __restrict__ x, const unsigned char* __restrict__ mask,
    const float* __restrict__ weight, const float* __restrict__ bias,
    const float* __restrict__ cm, const float* __restrict__ cM2,
    const float* __restrict__ ccnt, float* __restrict__ out) {
  __shared__ __align__(16) float ring[RING][D_DIM];
  const int b = blockIdx.x / NCHUNK;
  const int c = blockIdx.x % NCHUNK;
  const int tid = threadIdx.x;
  const int d4 = tid * 4;
  const int t0 = c * CHUNK;
  const size_t trow = (size_t)B_DIM * D_DIM;
  const float* xbase = x + (size_t)t0 * trow + (size_t)b * D_DIM + d4;
  float* obase = out + (size_t)t0 * trow + (size_t)b * D_DIM + d4;
  const unsigned char* mbase = mask + (size_t)b * T_DIM + t0;

  const size_t co = ((size_t)b * NCHUNK + c) * D_DIM + d4;
  float m0 = cm[co + 0], m1 = cm[co + 1], m2 = cm[co + 2], m3 = cm[co + 3];
  float s0 = cM2[co + 0], s1 = cM2[co + 1], s2 = cM2[co + 2], s3 = cM2[co + 3];
  float n = ccnt[b * NCHUNK + c];
  const float g0 = weight[d4 + 0] + 1.f, g1 = weight[d4 + 1] + 1.f;
  const float g2 = weight[d4 + 2] + 1.f, g3 = weight[d4 + 3] + 1.f;
  const float e0 = bias[d4 + 0], e1 = bias[d4 + 1];
  const float e2 = bias[d4 + 2], e3 = bias[d4 + 3];

  for (int i = 0; i < PF; ++i)
    async_load_b128(lds_addr_of(&ring[i][d4]), xbase + (size_t)i * trow);

  auto body = [&](int i) {
    const float4 v = *reinterpret_cast<const float4*>(&ring[i % RING][d4]);
    const float valid = mbase[i] ? 0.f : 1.f;
    const float n_new = n + valid;
    const float coef = (valid > 0.f) ? (valid / n_new) : 0.f;
    const float invn = 1.0f / n_new;   // n_new >= PRIOR_COUNT > 0 always
    float d;
    float4 y;
    d = v.x - m0; m0 += coef * d; s0 += valid * d * (v.x - m0);
    y.x = (v.x - m0) * rsqrtf(s0 * invn + EPS_F) * g0 + e0;
    d = v.y - m1; m1 += coef * d; s1 += valid * d * (v.y - m1);
    y.y = (v.y - m1) * rsqrtf(s1 * invn + EPS_F) * g1 + e1;
    d = v.z - m2; m2 += coef * d; s2 += valid * d * (v.z - m2);
    y.z = (v.z - m2) * rsqrtf(s2 * invn + EPS_F) * g2 + e2;
    d = v.w - m3; m3 += coef * d; s3 += valid * d * (v.w - m3);
    y.w = (v.w - m3) * rsqrtf(s3 * invn + EPS_F) * g3 + e3;
    n = n_new;
    *reinterpret_cast<float4*>(obase + (size_t)i * trow) = y;
  };

  for (int i = 0; i < CHUNK - PF; ++i) {
    S_WAIT_ASYNC_LE7();
    body(i);
    const int j = i + PF;
    async_load_b128(lds_addr_of(&ring[j % RING][d4]), xbase + (size_t)j * trow);
  }
  S_WAIT_ASYNC_0();
  for (int i = CHUNK - PF; i < CHUNK; ++i) body(i);
}

extern "C" void kernel_launch(void* const* d_in, const int* in_sizes, int n_in,
                              void* d_out, int out_size, void* d_ws, size_t ws_size,
                              hipStream_t stream) {
  const float* x          = (const float*)d_in[0];
  const float* prior_mean = (const float*)d_in[1];
  const float* prior_logv = (const float*)d_in[2];
  const float* weight     = (const float*)d_in[3];
  const float* bias       = (const float*)d_in[4];
  const unsigned char* pm = (const unsigned char*)d_in[5]; // bool mask (B,T)
  float* out = (float*)d_out;

  // Workspace layout (floats): ~2.0 MiB total.
  float* ws = (float*)d_ws;
  const size_t BCD = (size_t)B_DIM * NCHUNK * D_DIM;
  float* lm   = ws;
  float* lM2  = lm + BCD;
  float* cm   = lM2 + BCD;
  float* cM2  = cm + BCD;
  float* lcnt = cM2 + BCD;
  float* ccnt = lcnt + (size_t)B_DIM * NCHUNK;

  tsnorm_ph1<<<B_DIM * NCHUNK, TPB, 0, stream>>>(x, pm, lm, lM2, lcnt);
  tsnorm_ph2<<<B_DIM, D_DIM, 0, stream>>>(prior_mean, prior_logv, lm, lM2, lcnt,
                                          cm, cM2, ccnt);
  tsnorm_ph3<<<B_DIM * NCHUNK, TPB, 0, stream>>>(x, pm, weight, bias,
                                                 cm, cM2, ccnt, out);
}